// Model_23751169147047
// MI455X (gfx1250) — compile-verified
//
#include <hip/hip_runtime.h>
#include <cstdint>
#include <cstddef>

#define USER_N   100000
#define ITEM_N   200000
#define NN       (USER_N + ITEM_N)   // 300000
#define DIM      64
#define E_EDGES  9600000
#define NLAYERS  3
#define TILE     256                 // edges per LDS tile
#define TPB_TILES 4                  // tiles per block (double-buffered)
// E_EDGES = 256 * 37500 ; 37500 / 4 = 9375 blocks exactly

// ---------------- CDNA5 async global->LDS staging -------------------------
#if defined(__gfx1250__) && __has_builtin(__builtin_amdgcn_global_load_async_to_lds_b32)
#define HAVE_ASYNC_LDS 1
#else
#define HAVE_ASYNC_LDS 0
#endif

#if HAVE_ASYNC_LDS
typedef __attribute__((address_space(1))) int as1_int;
typedef __attribute__((address_space(3))) int as3_int;

__device__ __forceinline__ void async_copy_b32(const void* gsrc, void* lds_dst) {
  __builtin_amdgcn_global_load_async_to_lds_b32(
      (as1_int*)gsrc, (as3_int*)lds_dst, /*offset=*/0, /*cpol=*/0);
}
#endif

template <int N>
__device__ __forceinline__ void wait_async_le() {
#if HAVE_ASYNC_LDS
#if __has_builtin(__builtin_amdgcn_s_wait_asynccnt)
  __builtin_amdgcn_s_wait_asynccnt(N);
#else
  asm volatile("s_wait_asynccnt %0" ::"i"(N) : "memory");
#endif
#endif
}

// ---------------- SpMM: y[row] += val * x[col]  (y pre-zeroed) ------------
// Block owns TPB_TILES consecutive tiles of 256 edges. Edge (row,col,val)
// streams are staged global->LDS with async DMA, double-buffered so tile k+1
// is in flight while tile k is processed. 16 threads per edge; each lane
// handles one float4 slice of DIM=64: 256B coalesced gather + 4 f32 atomics.
__global__ __launch_bounds__(256) void spmm_kernel(
    const int* __restrict__ erow, const int* __restrict__ ecol,
    const float* __restrict__ eval,
    const float* __restrict__ x, float* __restrict__ y) {
  __shared__ int   s_row[2][TILE];
  __shared__ int   s_col[2][TILE];
  __shared__ float s_val[2][TILE];

  const int t = threadIdx.x;
  const long base0 = (long)blockIdx.x * (TILE * TPB_TILES);
  const int g = t >> 4;   // edge group 0..15
  const int l = t & 15;   // feature chunk 0..15 (float4 each)

  // stage tile 0 into buffer 0
#if HAVE_ASYNC_LDS
  async_copy_b32(erow + base0 + t, &s_row[0][t]);
  async_copy_b32(ecol + base0 + t, &s_col[0][t]);
  async_copy_b32(eval + base0 + t, &s_val[0][t]);
#else
  s_row[0][t] = erow[base0 + t];
  s_col[0][t] = ecol[base0 + t];
  s_val[0][t] = eval[base0 + t];
#endif

  int cur = 0;
#pragma unroll
  for (int k = 0; k < TPB_TILES; ++k) {
    // kick off the next tile into the other buffer
    if (k + 1 < TPB_TILES) {
      const long nb = base0 + (long)(k + 1) * TILE;
#if HAVE_ASYNC_LDS
      async_copy_b32(erow + nb + t, &s_row[cur ^ 1][t]);
      async_copy_b32(ecol + nb + t, &s_col[cur ^ 1][t]);
      async_copy_b32(eval + nb + t, &s_val[cur ^ 1][t]);
#else
      s_row[cur ^ 1][t] = erow[nb + t];
      s_col[cur ^ 1][t] = ecol[nb + t];
      s_val[cur ^ 1][t] = eval[nb + t];
#endif
      // <=3 outstanding => this wave's tile-k loads (older, in-order) are done
      wait_async_le<3>();
    } else {
      wait_async_le<0>();
    }
    __syncthreads();   // all waves' tile-k data now visible in LDS

#pragma unroll
    for (int j = 0; j < 16; ++j) {
      const int   el = g + (j << 4);        // conflict-free LDS access
      const int   r  = s_row[cur][el];
      const int   c  = s_col[cur][el];
      const float v  = s_val[cur][el];

      const float4 xv =
          *reinterpret_cast<const float4*>(x + (size_t)c * DIM + (l << 2));
      float* yp = y + (size_t)r * DIM + (l << 2);

      unsafeAtomicAdd(yp + 0, xv.x * v);   // global_atomic_add_f32, no RTN
      unsafeAtomicAdd(yp + 1, xv.y * v);
      unsafeAtomicAdd(yp + 2, xv.z * v);
      unsafeAtomicAdd(yp + 3, xv.w * v);
    }
    __syncthreads();   // done reading buf[cur] before it is restaged
    cur ^= 1;
  }
}

// ---------------- elementwise kernels -------------------------------------

// out = cur = concat(u, i); zero first scatter buffer (float4 indexed)
__global__ __launch_bounds__(256) void init_kernel(
    const float4* __restrict__ u, const float4* __restrict__ it,
    float4* __restrict__ out, float4* __restrict__ cur,
    float4* __restrict__ zbuf) {
  const int i = blockIdx.x * 256 + threadIdx.x;      // < NN*DIM/4
  constexpr int U4 = USER_N * DIM / 4;
  const float4 v = (i < U4) ? u[i] : it[i - U4];
  out[i] = v;
  cur[i] = v;
  zbuf[i] = make_float4(0.f, 0.f, 0.f, 0.f);
}

// out += src; also zero the dead ping-pong buffer for the next layer
__global__ __launch_bounds__(256) void add_zero_kernel(
    const float4* __restrict__ src, float4* __restrict__ out,
    float4* __restrict__ zbuf) {
  const int i = blockIdx.x * 256 + threadIdx.x;
  float4 a = out[i];
  const float4 b = src[i];
  a.x += b.x; a.y += b.y; a.z += b.z; a.w += b.w;
  out[i] = a;
  zbuf[i] = make_float4(0.f, 0.f, 0.f, 0.f);
}

// ---------------- launch --------------------------------------------------

extern "C" void kernel_launch(void* const* d_in, const int* in_sizes, int n_in,
                              void* d_out, int out_size, void* d_ws, size_t ws_size,
                              hipStream_t stream) {
  const float* u    = (const float*)d_in[0];
  const float* it   = (const float*)d_in[1];
  const int*   erow = (const int*)d_in[2];
  const int*   ecol = (const int*)d_in[3];
  const float* ev   = (const float*)d_in[4];
  float* out = (float*)d_out;

  float* bufA = (float*)d_ws;                 // current layer input
  float* bufB = bufA + (size_t)NN * DIM;      // scatter target (zeroed)

  const int n4      = NN * DIM / 4;           // 4,800,000
  const int nBlocks = n4 / 256;               // 18,750 exactly
  const int spmmBlocks = E_EDGES / (TILE * TPB_TILES);  // 9,375 exactly

  init_kernel<<<nBlocks, 256, 0, stream>>>(
      (const float4*)u, (const float4*)it,
      (float4*)out, (float4*)bufA, (float4*)bufB);

  for (int layer = 0; layer < NLAYERS; ++layer) {
    spmm_kernel<<<spmmBlocks, 256, 0, stream>>>(erow, ecol, ev, bufA, bufB);
    // out += bufB ; zero bufA (it becomes next layer's scatter target)
    add_zero_kernel<<<nBlocks, 256, 0, stream>>>(
        (const float4*)bufB, (float4*)out, (float4*)bufA);
    float* tmp = bufA; bufA = bufB; bufB = tmp;
  }
}